// LSTMModel_16166256902522
// MI455X (gfx1250) — compile-verified
//
#include <hip/hip_runtime.h>

typedef __attribute__((ext_vector_type(2))) float v2f;
typedef __attribute__((ext_vector_type(8))) float v8f;

// Problem constants (from reference): B,T,D,H,O = 4096,512,6,30,10
constexpr int Bsz = 4096;
constexpr int T   = 512;
constexpr int Din = 6;
constexpr int H   = 30;
constexpr int G   = 4 * H;     // 120 gate columns
constexpr int O   = 10;
constexpr int NP  = 128;       // gate columns padded to 8 WMMA n-tiles
constexpr int KT0 = 10;        // layer0 K = 40  : [h(0..29) | pad | x(32..37) | pad]
constexpr int K0  = 4 * KT0;
constexpr int KT1 = 16;        // layer1 K = 64  : [h2(0..29) | pad | h1(32..61) | pad]
constexpr int K1  = 4 * KT1;

// CDNA5 hardware tanh (single TRANS op). ISA §7.4: a TRANS op needs one
// independent op or V_NOP before its result is consumed; the compiler can't
// see through the asm, so we pay the v_nop explicitly.
static __device__ __forceinline__ float htanh(float x) {
  float r;
  asm volatile("v_tanh_f32 %0, %1\n\tv_nop" : "=v"(r) : "v"(x));
  return r;
}
// sigmoid(x) = 0.5*tanh(x/2) + 0.5  -> one v_tanh + fma
static __device__ __forceinline__ float hsig(float x) {
  return __builtin_fmaf(0.5f, htanh(0.5f * x), 0.5f);
}

static __device__ __forceinline__ v8f bcast8(float v) {
  v8f r;
  r[0]=v; r[1]=v; r[2]=v; r[3]=v; r[4]=v; r[5]=v; r[6]=v; r[7]=v;
  return r;
}

__global__ __launch_bounds__(128)
void lstm2_fused_wmma(const float* __restrict__ x,
                      const float* __restrict__ Wih0, const float* __restrict__ Whh0,
                      const float* __restrict__ bih0, const float* __restrict__ bhh0,
                      const float* __restrict__ Wih1, const float* __restrict__ Whh1,
                      const float* __restrict__ bih1, const float* __restrict__ bhh1,
                      const float* __restrict__ Wout, const float* __restrict__ bout,
                      float* __restrict__ out)
{
  // Weight fragments pre-packed for v_wmma_f32_16x16x4_f32 B-operand:
  // lane<16 holds rows {4kt+0, 4kt+1}, lane>=16 holds rows {4kt+2, 4kt+3}.
  __shared__ v2f   Wp0[KT0][2][NP];   // 20 KB
  __shared__ v2f   Wp1[KT1][2][NP];   // 32 KB
  __shared__ float gbuf[16][NP];      //  8 KB  (gate matrix staging)
  __shared__ float abufA[16][K0];     // 2.5 KB (layer0 A rows: [h0 | x_t])
  __shared__ float abufB[16][K1];     //  4 KB  (layer1 A rows: [h2 | h1_t])

  const int tid  = threadIdx.x;
  const int wv   = tid >> 5;          // wave 0..3 -> n-tiles {2wv, 2wv+1}
  const int lane = tid & 31;
  const int ln   = lane & 15;
  const int hi   = lane >> 4;
  const long bbase = (long)blockIdx.x * 16;

  // ---------------- pack weights into LDS (once) ----------------
  // Layer0 combined: W0[k][n] = k<30 ? Whh0[n][k] : (32<=k<38 ? Wih0[n][k-32] : 0)
  for (int idx = tid; idx < KT0 * 2 * NP; idx += 128) {
    int n    = idx & (NP - 1);
    int pair = idx >> 7;               // 0..19
    int kt   = pair >> 1;
    int hh   = pair & 1;
    int k0   = kt * 4 + 2 * hh;
    float a = 0.f, b = 0.f;
    if (n < G) {
      int k = k0;
      a = (k < H) ? Whh0[n * H + k]
        : ((k >= 32 && k < 32 + Din) ? Wih0[n * Din + (k - 32)] : 0.f);
      k = k0 + 1;
      b = (k < H) ? Whh0[n * H + k]
        : ((k >= 32 && k < 32 + Din) ? Wih0[n * Din + (k - 32)] : 0.f);
    }
    v2f w; w.x = a; w.y = b;
    Wp0[kt][hh][n] = w;
  }
  // Layer1 combined: W1[k][n] = k<30 ? Whh1[n][k] : (32<=k<62 ? Wih1[n][k-32] : 0)
  for (int idx = tid; idx < KT1 * 2 * NP; idx += 128) {
    int n    = idx & (NP - 1);
    int pair = idx >> 7;               // 0..31
    int kt   = pair >> 1;
    int hh   = pair & 1;
    int k0   = kt * 4 + 2 * hh;
    float a = 0.f, b = 0.f;
    if (n < G) {
      int k = k0;
      a = (k < H) ? Whh1[n * H + k]
        : ((k >= 32 && k < 32 + H) ? Wih1[n * H + (k - 32)] : 0.f);
      k = k0 + 1;
      b = (k < H) ? Whh1[n * H + k]
        : ((k >= 32 && k < 32 + H) ? Wih1[n * H + (k - 32)] : 0.f);
    }
    v2f w; w.x = a; w.y = b;
    Wp1[kt][hh][n] = w;
  }
  // Zero A staging (pad slots stay zero forever; h/x slots get overwritten)
  for (int idx = tid; idx < 16 * K0; idx += 128) abufA[idx / K0][idx % K0] = 0.f;
  for (int idx = tid; idx < 16 * K1; idx += 128) abufB[idx / K1][idx % K1] = 0.f;

  // Per-lane gate biases for this wave's two n-tiles (folded into WMMA C init)
  float bias0a, bias0b, bias1a, bias1b;
  {
    int n0 = (wv * 2 + 0) * 16 + ln;
    int n1 = (wv * 2 + 1) * 16 + ln;
    bias0a = (n0 < G) ? (bih0[n0] + bhh0[n0]) : 0.f;
    bias0b = (n1 < G) ? (bih0[n1] + bhh0[n1]) : 0.f;
    bias1a = (n0 < G) ? (bih1[n0] + bhh1[n0]) : 0.f;
    bias1b = (n1 < G) ? (bih1[n1] + bhh1[n1]) : 0.f;
  }

  // Cell state kept in registers: lane owns (m,j) pairs p = tid + 128*q
  float c0q[4] = {0.f, 0.f, 0.f, 0.f};
  float c1q[4] = {0.f, 0.f, 0.f, 0.f};

  __syncthreads();

  const int m = ln;  // WMMA A row owned by this lane

  for (int t = 0; t < T; ++t) {
    // ---- stage x_t into layer0 A rows (slots 32..37) ----
    if (tid < 16 * Din) {
      int mm = tid / Din, d = tid % Din;
      abufA[mm][32 + d] = x[((bbase + mm) * T + t) * Din + d];
    }
    if (t + 1 < T && tid < 16) {
      __builtin_prefetch(&x[((bbase + tid) * T + (t + 1)) * Din], 0, 0);
    }
    __syncthreads();

    // =============== layer 0: gates = [h0|x] @ W0 + bias ===============
    {
      v8f acc0 = bcast8(bias0a);
      v8f acc1 = bcast8(bias0b);
#pragma unroll
      for (int kt = 0; kt < KT0; ++kt) {
        v2f a;
        a.x = abufA[m][kt * 4 + 2 * hi + 0];
        a.y = abufA[m][kt * 4 + 2 * hi + 1];
        v2f bA = Wp0[kt][hi][(wv * 2 + 0) * 16 + ln];
        v2f bB = Wp0[kt][hi][(wv * 2 + 1) * 16 + ln];
        acc0 = __builtin_amdgcn_wmma_f32_16x16x4_f32(false, a, false, bA,
                                                     (short)0, acc0, false, false);
        acc1 = __builtin_amdgcn_wmma_f32_16x16x4_f32(false, a, false, bB,
                                                     (short)0, acc1, false, false);
      }
#pragma unroll
      for (int r = 0; r < 8; ++r) {
        gbuf[r + 8 * hi][(wv * 2 + 0) * 16 + ln] = acc0[r];
        gbuf[r + 8 * hi][(wv * 2 + 1) * 16 + ln] = acc1[r];
      }
    }
    __syncthreads();

    // ---- layer 0 cell update (hardware v_tanh_f32 path) ----
#pragma unroll
    for (int q = 0; q < 4; ++q) {
      int p = tid + 128 * q;
      if (p < 16 * H) {
        int mm = p & 15, j = p >> 4;
        float ig = gbuf[mm][j];
        float fg = gbuf[mm][H + j];
        float gg = gbuf[mm][2 * H + j];
        float og = gbuf[mm][3 * H + j];
        float c  = hsig(fg) * c0q[q] + hsig(ig) * htanh(gg);
        c0q[q]   = c;
        float hn = hsig(og) * htanh(c);
        abufA[mm][j]      = hn;   // recurrent input for next layer0 step
        abufB[mm][32 + j] = hn;   // h1_t feeds layer1 now
      }
    }
    __syncthreads();

    // =============== layer 1: gates = [h2|h1] @ W1 + bias ===============
    {
      v8f acc0 = bcast8(bias1a);
      v8f acc1 = bcast8(bias1b);
#pragma unroll
      for (int kt = 0; kt < KT1; ++kt) {
        v2f a;
        a.x = abufB[m][kt * 4 + 2 * hi + 0];
        a.y = abufB[m][kt * 4 + 2 * hi + 1];
        v2f bA = Wp1[kt][hi][(wv * 2 + 0) * 16 + ln];
        v2f bB = Wp1[kt][hi][(wv * 2 + 1) * 16 + ln];
        acc0 = __builtin_amdgcn_wmma_f32_16x16x4_f32(false, a, false, bA,
                                                     (short)0, acc0, false, false);
        acc1 = __builtin_amdgcn_wmma_f32_16x16x4_f32(false, a, false, bB,
                                                     (short)0, acc1, false, false);
      }
#pragma unroll
      for (int r = 0; r < 8; ++r) {
        gbuf[r + 8 * hi][(wv * 2 + 0) * 16 + ln] = acc0[r];
        gbuf[r + 8 * hi][(wv * 2 + 1) * 16 + ln] = acc1[r];
      }
    }
    __syncthreads();

    // ---- layer 1 cell update ----
#pragma unroll
    for (int q = 0; q < 4; ++q) {
      int p = tid + 128 * q;
      if (p < 16 * H) {
        int mm = p & 15, j = p >> 4;
        float ig = gbuf[mm][j];
        float fg = gbuf[mm][H + j];
        float gg = gbuf[mm][2 * H + j];
        float og = gbuf[mm][3 * H + j];
        float c  = hsig(fg) * c1q[q] + hsig(ig) * htanh(gg);
        c1q[q]   = c;
        abufB[mm][j] = hsig(og) * htanh(c);   // h2 recurrent slot (also final state)
      }
    }
    __syncthreads();
  }

  // ---------------- output: h2_last @ Wout.T + bout ----------------
  for (int p = tid; p < 16 * O; p += 128) {
    int mm = p & 15, o = p >> 4;
    float v = bout[o];
#pragma unroll
    for (int j = 0; j < H; ++j) v += abufB[mm][j] * Wout[o * H + j];
    out[(bbase + mm) * O + o] = v;
  }
}

extern "C" void kernel_launch(void* const* d_in, const int* in_sizes, int n_in,
                              void* d_out, int out_size, void* d_ws, size_t ws_size,
                              hipStream_t stream) {
  (void)in_sizes; (void)n_in; (void)d_ws; (void)ws_size; (void)out_size;
  const float* x    = (const float*)d_in[0];
  const float* Wih0 = (const float*)d_in[1];
  const float* Whh0 = (const float*)d_in[2];
  const float* bih0 = (const float*)d_in[3];
  const float* bhh0 = (const float*)d_in[4];
  const float* Wih1 = (const float*)d_in[5];
  const float* Whh1 = (const float*)d_in[6];
  const float* bih1 = (const float*)d_in[7];
  const float* bhh1 = (const float*)d_in[8];
  const float* Wout = (const float*)d_in[9];
  const float* bout = (const float*)d_in[10];
  float* out = (float*)d_out;

  dim3 grid(Bsz / 16), block(128);
  hipLaunchKernelGGL(lstm2_fused_wmma, grid, block, 0, stream,
                     x, Wih0, Whh0, bih0, bhh0, Wih1, Whh1, bih1, bhh1,
                     Wout, bout, out);
}